// CindexLoss_91036126806432
// MI455X (gfx1250) — compile-verified
//
#include <hip/hip_runtime.h>

// CDNA5 / gfx1250 implementation of the pairwise concordance loss.
//
// Key identity: flag_i*flag_j*(p_i-p_j)*(t_i-t_j) is a RANK-4 bilinear form:
//   A_i = f_i * [p_i*t_i, -p_i, -t_i, 1]   (K = 0..3)
//   B_j = f_j * [1, t_j, p_j, p_j*t_j]     (K = 0..3)
// so each 16x16 tile of the 8192x8192 pair matrix is ONE
// v_wmma_f32_16x16x4_f32 (exact fp32 math), followed by per-element relu
// and accumulation. relu(m*x) = m*relu(x) for m>=0 lets the 0/1 flags and
// the /100 scale be folded in/out of the matmul.
//
// Inputs are 96 KB -> fully L2 resident (192 MB L2); compute-shaped, so the
// goal is a minimal branch-free WMMA inner loop. The diagonal tile (strict
// upper-triangle mask) occurs at most once per wave and is peeled out.

typedef __attribute__((ext_vector_type(2))) float v2f;
typedef __attribute__((ext_vector_type(8))) float v8f;

// Single-instruction relu: v_max_num_f32 is IEEE maximumNumber (NaN-aware),
// avoiding the canonicalize+max pair that strict fmaxf lowering emits.
__device__ __forceinline__ float relu1(float x) {
    float r;
    asm("v_max_num_f32 %0, %1, 0" : "=v"(r) : "v"(x));
    return r;
}

// One 16x16 pair tile: build B operand for column block jb, multiply by the
// wave's resident A operand. B layout (4x16 f32, 2 VGPRs, mirrors the ISA's
// 16x4 A table): lanes 0-15 hold K0 (x) / K1 (y) for col N=lane; lanes
// 16-31 hold K2 (x) / K3 (y).
__device__ __forceinline__ v8f pair_tile(
    v2f A, const float* __restrict__ pred, const float2* __restrict__ gt2,
    int jb, int r16, bool hi, int col0)
{
    const int    gj = jb * 16 + r16;
    const float  pj = pred[gj];                 // global_load_b32
    const float2 g  = gt2[gj];                  // global_load_b64 (both cols)
    const float  tj = (col0 == 0) ? g.x : g.y;  // fracTime
    const float  mj = (col0 == 0) ? g.y : g.x;  // ifMOF
    const float  fj = (mj == 1.0f) ? 1.0f : 0.0f;
    v2f Bv;
    Bv.x = hi ? (fj * pj)      : fj;            // K2 : K0
    Bv.y = hi ? (fj * pj * tj) : (fj * tj);     // K3 : K1
    v8f cz = {0.f, 0.f, 0.f, 0.f, 0.f, 0.f, 0.f, 0.f};
    return __builtin_amdgcn_wmma_f32_16x16x4_f32(
        /*neg_a=*/false, A, /*neg_b=*/false, Bv,
        /*c_mod=*/(short)0, cz, /*reuse_a=*/false, /*reuse_b=*/false);
}

// grid = (NB16, 2), block = 256 (8 waves). Wave handles row-block ib =
// blockIdx.x and column-block chunk c = blockIdx.y*8 + waveId, covering
// jb in [c*chunkW, (c+1)*chunkW) clipped to the upper triangle (jb >= ib).
__global__ __launch_bounds__(256) void cindex_pairs_kernel(
    const float* __restrict__ pred, const float* __restrict__ gt,
    const int* __restrict__ colT, const int* __restrict__ colF,
    float* __restrict__ partial, int NB16, int chunkW)
{
    const int ib   = blockIdx.x;
    const int wave = threadIdx.x >> 5;
    const int lane = threadIdx.x & 31;
    const int c    = blockIdx.y * 8 + wave;   // chunk id 0..15
    const int slot = ib * 16 + c;             // unique deterministic slot

    const int col0 = *colT;  // fracTime column index (0)
    const float2* gt2 = (const float2*)gt;
    (void)colF;              // column roles derived from col0 (2 columns)

    const int jb0   = c * chunkW;
    int       jbEnd = jb0 + chunkW;
    if (jbEnd > NB16) jbEnd = NB16;

    const int  r16 = lane & 15;
    const bool hi  = lane >= 16;

    // A operand (16x4 f32, 2 VGPRs): per ISA table, lanes 0-15 hold K=0 (x)
    // and K=1 (y) for row M=lane; lanes 16-31 hold K=2 (x) and K=3 (y).
    const int    gi = ib * 16 + r16;
    const float  pi = pred[gi];
    const float2 gA = gt2[gi];
    const float  ti = (col0 == 0) ? gA.x : gA.y;
    const float  mi = (col0 == 0) ? gA.y : gA.x;
    const float  fi = (mi == 1.0f) ? 1.0f : 0.0f;
    v2f A;
    A.x = hi ? (-fi * ti) : (fi * pi * ti);   // K2 : K0
    A.y = hi ? fi         : (-fi * pi);       // K3 : K1

    v8f acc = {0.f, 0.f, 0.f, 0.f, 0.f, 0.f, 0.f, 0.f};

    // ---- Peeled diagonal tile (at most one per wave) -------------------
    int jb = jb0;
    if (ib >= jbEnd) {
        jb = jbEnd;                           // chunk entirely below diagonal
    } else if (ib >= jb0) {
        // Diagonal tile jb == ib: keep strict upper triangle only.
        // C/D layout: element (M = r + 8*(lane>=16), N = lane&15).
        // Mask BEFORE the relu so the select is a speculatable v_cndmask
        // and the asm relu stays unconditional (no EXEC branching):
        // relu(mask ? d : 0) == mask01 * relu(d) since relu(0) == 0.
        v8f d = pair_tile(A, pred, gt2, ib, r16, hi, col0);
        const int Mbase = hi ? 8 : 0;
#pragma unroll
        for (int r = 0; r < 8; ++r) {
            const float dm = ((Mbase + r) < r16) ? d[r] : 0.f;
            acc[r] += relu1(dm);
        }
        jb = ib + 1;
    }

    // ---- Branch-free hot loop: full off-diagonal tiles -----------------
    for (; jb < jbEnd; ++jb) {
        v8f d = pair_tile(A, pred, gt2, jb, r16, hi, col0);
#pragma unroll
        for (int r = 0; r < 8; ++r)
            acc[r] += relu1(d[r]);            // v_max_num_f32 + v_add_f32
    }

    // Per-lane then cross-lane (wave32) reduction; lane 0 writes the slot.
    float s = acc[0] + acc[1] + acc[2] + acc[3] +
              acc[4] + acc[5] + acc[6] + acc[7];
#pragma unroll
    for (int off = 16; off > 0; off >>= 1)
        s += __shfl_xor(s, off, 32);
    if (lane == 0) partial[slot] = s;   // every wave writes -> no init pass
}

// ---- Finalize: deterministic fixed-order reduction + denominator -------
__global__ __launch_bounds__(256) void cindex_finalize_kernel(
    const float* __restrict__ gt, const int* __restrict__ colF,
    const float* __restrict__ partial, float* __restrict__ out,
    int Bn, int nslots)
{
    __shared__ float sm[256];
    __shared__ float sf[256];
    const int tid  = threadIdx.x;
    const int col1 = *colF;

    float s = 0.f, fl = 0.f;
    for (int i = tid; i < nslots; i += 256) s += partial[i];
    for (int i = tid; i < Bn; i += 256)
        fl += (gt[i * 2 + col1] == 1.0f) ? 1.0f : 0.0f;

    sm[tid] = s;
    sf[tid] = fl;
    __syncthreads();
    for (int st = 128; st > 0; st >>= 1) {
        if (tid < st) { sm[tid] += sm[tid + st]; sf[tid] += sf[tid + st]; }
        __syncthreads();
    }
    if (tid == 0) {
        const float S      = sf[0];
        const float npairs = 0.5f * S * (S - 1.0f);  // flagged pair count
        out[0] = (sm[0] * 0.01f) / npairs;           // fold the /100 here
    }
}

extern "C" void kernel_launch(void* const* d_in, const int* in_sizes, int n_in,
                              void* d_out, int out_size, void* d_ws, size_t ws_size,
                              hipStream_t stream) {
    const float* pred = (const float*)d_in[0];
    const float* gt   = (const float*)d_in[1];
    const int*   colT = (const int*)d_in[2];   // gt_fracTime (=0)
    const int*   colF = (const int*)d_in[3];   // gt_ifMOF    (=1)
    float* out     = (float*)d_out;
    float* partial = (float*)d_ws;             // NB16*16 floats (32 KB @ B=8192)

    const int Bn     = in_sizes[0];            // 8192
    const int NB16   = Bn / 16;                // 512 row/col blocks
    const int chunkW = (NB16 + 15) / 16;       // 32 column blocks per wave
    const int nslots = NB16 * 16;              // 8192 partials

    dim3 grid(NB16, 2);
    cindex_pairs_kernel<<<grid, 256, 0, stream>>>(pred, gt, colT, colF,
                                                  partial, NB16, chunkW);
    cindex_finalize_kernel<<<1, 256, 0, stream>>>(gt, colF, partial, out,
                                                  Bn, nslots);
}